// AttentionLayer_15144054685952
// MI455X (gfx1250) — compile-verified
//
#include <hip/hip_runtime.h>
#include <hip/hip_bf16.h>
#include <math.h>

// Problem constants (from reference)
#define BB  16
#define S1  128
#define S2  256
#define HH  256
#define HM  512

typedef __attribute__((ext_vector_type(2))) float v2f;
typedef __attribute__((ext_vector_type(8))) float v8f;

// ---------------------------------------------------------------------------
// Generic batched fp32 GEMM via V_WMMA_F32_16X16X4_F32.
//   Y[b] = X[b] @ W[b] (+ bias), row-major. ldX = K, ldW = N, ldY = N.
//   Block = 256 threads (8 waves). Block tile = 128 (M) x 16 (N).
//   K must be a multiple of 64; M mult of 128; N mult of 16.
// ---------------------------------------------------------------------------
__global__ __launch_bounds__(256) void wmma_gemm_f32(
    const float* __restrict__ X, const float* __restrict__ W,
    const float* __restrict__ bias, float* __restrict__ Y,
    int M, int N, int K,
    long long strideX, long long strideW, long long strideY)
{
    const int lane = threadIdx.x & 31;
    const int wave = threadIdx.x >> 5;     // 0..7
    const int half = lane >> 4;            // 0 or 1
    const int row  = lane & 15;            // 0..15
    const int tileN  = blockIdx.x * 16;
    const int tileM0 = blockIdx.y * 128;

    const float* Xb = X + (long long)blockIdx.z * strideX;
    const float* Wb = W + (long long)blockIdx.z * strideW;
    float*       Yb = Y + (long long)blockIdx.z * strideY;

    const int rowM = tileM0 + wave * 16 + row;      // A-row this lane feeds

    __shared__ float Bs[64][16];                    // K-panel of W

    v8f acc = {};

    for (int k0 = 0; k0 < K; k0 += 64) {
        // Cooperative load of 64x16 B panel (1024 floats / 256 threads = 4 each)
        {
            const int t  = threadIdx.x;
            const int kk = t >> 2;                  // 0..63
            const int nn = (t & 3) << 2;            // 0,4,8,12
            const float4 v = *(const float4*)(Wb + (long long)(k0 + kk) * N + tileN + nn);
            Bs[kk][nn + 0] = v.x;
            Bs[kk][nn + 1] = v.y;
            Bs[kk][nn + 2] = v.z;
            Bs[kk][nn + 3] = v.w;
        }
        __syncthreads();

        const float* ap = Xb + (long long)rowM * K + k0 + half * 2;
        #pragma unroll
        for (int ks = 0; ks < 64; ks += 4) {
            v2f a, b;
            a.x = ap[ks + 0];
            a.y = ap[ks + 1];
            b.x = Bs[ks + half * 2 + 0][row];
            b.y = Bs[ks + half * 2 + 1][row];
            // D = A(16x4) * B(4x16) + C
            acc = __builtin_amdgcn_wmma_f32_16x16x4_f32(
                false, a, false, b, (short)0, acc, false, false);
        }
        __syncthreads();
    }

    const float bv = bias ? bias[tileN + row] : 0.0f;
    #pragma unroll
    for (int i = 0; i < 8; ++i) {
        const int m = tileM0 + wave * 16 + i + half * 8;   // C/D layout: VGPR i -> rows i, i+8
        Yb[(long long)m * N + tileN + row] = acc[i] + bv;
    }
}

// ---------------------------------------------------------------------------
// Fused score kernel: one block per (b,q); thread t owns k = t.
//   score = sum_h relu(qp[b,q,h] + kp[b,k,h]) * w2[h]  (+ b2)
//   A = mask ? exp(score) : 0 ; block-reduce row sum; write A / clip(sum).
// qp row and w2 staged in LDS (broadcast); kp streamed as float4 (L2-resident).
// ---------------------------------------------------------------------------
__global__ __launch_bounds__(256) void score_attn_kernel(
    const float* __restrict__ qp, const float* __restrict__ kp,
    const float* __restrict__ w2, const float* __restrict__ b2,
    const float* __restrict__ q_mask, const float* __restrict__ k_mask,
    float* __restrict__ attn)
{
    const int q = blockIdx.x;      // 0..S1-1
    const int b = blockIdx.y;      // 0..B-1
    const int k = threadIdx.x;     // 0..S2-1 (S2 == 256 == blockDim.x)

    __shared__ float qs[HM];
    __shared__ float w2s[HM];
    __shared__ float red[8];

    const float* qrow = qp + ((long long)b * S1 + q) * HM;
    qs[k]        = qrow[k];
    qs[k + 256]  = qrow[k + 256];
    w2s[k]       = w2[k];
    w2s[k + 256] = w2[k + 256];
    __syncthreads();

    const float* krow = kp + ((long long)b * S2 + k) * HM;
    float acc = 0.0f;
    #pragma unroll 4
    for (int h = 0; h < HM; h += 4) {
        const float4 kv = *(const float4*)(krow + h);
        acc = fmaf(fmaxf(qs[h + 0] + kv.x, 0.0f), w2s[h + 0], acc);
        acc = fmaf(fmaxf(qs[h + 1] + kv.y, 0.0f), w2s[h + 1], acc);
        acc = fmaf(fmaxf(qs[h + 2] + kv.z, 0.0f), w2s[h + 2], acc);
        acc = fmaf(fmaxf(qs[h + 3] + kv.w, 0.0f), w2s[h + 3], acc);
    }
    const float score = acc + b2[0];
    const float m = q_mask[(long long)b * S1 + q] * k_mask[(long long)b * S2 + k];
    const float A = (m == 0.0f) ? 0.0f : expf(score);

    // block reduction of A over 256 threads (8 waves of 32)
    const int lane = threadIdx.x & 31;
    const int wave = threadIdx.x >> 5;
    float s = A;
    #pragma unroll
    for (int off = 16; off > 0; off >>= 1) s += __shfl_down(s, off, 32);
    if (lane == 0) red[wave] = s;
    __syncthreads();
    float total = 0.0f;
    #pragma unroll
    for (int i = 0; i < 8; ++i) total += red[i];
    total = fmaxf(total, 2e-15f);

    attn[((long long)b * S1 + q) * S2 + k] = A / total;
}

// ---------------------------------------------------------------------------
// Launch: qp-GEMM, kp-GEMM, fused score/softmax, attn@value batched GEMM.
// ---------------------------------------------------------------------------
extern "C" void kernel_launch(void* const* d_in, const int* in_sizes, int n_in,
                              void* d_out, int out_size, void* d_ws, size_t ws_size,
                              hipStream_t stream)
{
    const float* query  = (const float*)d_in[0];   // [B,S1,H]
    const float* key    = (const float*)d_in[1];   // [B,S2,H]
    const float* value  = (const float*)d_in[2];   // [B,S2,H]
    const float* q_mask = (const float*)d_in[3];   // [B,S1]
    const float* k_mask = (const float*)d_in[4];   // [B,S2]
    const float* W1     = (const float*)d_in[5];   // [2H, HM]
    const float* b1     = (const float*)d_in[6];   // [HM]
    const float* W2     = (const float*)d_in[7];   // [HM,1]
    const float* b2     = (const float*)d_in[8];   // [1]
    float* out = (float*)d_out;                    // [B,S1,H]

    float* qp_ws   = (float*)d_ws;                               // B*S1*HM
    float* kp_ws   = qp_ws + (size_t)BB * S1 * HM;               // B*S2*HM
    float* attn_ws = kp_ws + (size_t)BB * S2 * HM;               // B*S1*S2

    // 1) qp = query @ W1[:H] + b1   (b1 folded here once; hid = qp + kp + b1)
    wmma_gemm_f32<<<dim3(HM / 16, (BB * S1) / 128, 1), 256, 0, stream>>>(
        query, W1, b1, qp_ws, BB * S1, HM, HH, 0, 0, 0);

    // 2) kp = key @ W1[H:]
    wmma_gemm_f32<<<dim3(HM / 16, (BB * S2) / 128, 1), 256, 0, stream>>>(
        key, W1 + (size_t)HH * HM, nullptr, kp_ws, BB * S2, HM, HH, 0, 0, 0);

    // 3) fused relu/W2 contraction + mask + exp + rowsum + normalize
    score_attn_kernel<<<dim3(S1, BB, 1), 256, 0, stream>>>(
        qp_ws, kp_ws, W2, b2, q_mask, k_mask, attn_ws);

    // 4) out[b] = attn[b] @ value[b]   (batched: M=S1, N=H, K=S2)
    wmma_gemm_f32<<<dim3(HH / 16, S1 / 128, BB), 256, 0, stream>>>(
        attn_ws, value, nullptr, out, S1, HH, S2,
        (long long)S1 * S2, (long long)S2 * HH, (long long)S1 * HH);
}